// GCNLayer_30966714204803
// MI455X (gfx1250) — compile-verified
//
#include <hip/hip_runtime.h>

typedef __attribute__((ext_vector_type(2))) float v2f;
typedef __attribute__((ext_vector_type(8))) float v8f;
typedef int b128_t __attribute__((vector_size(16)));  // async-LDS builtin operand type

#define D 128  // D_IN == D_OUT == 128

// CDNA5 async global->LDS path (ASYNCcnt), if this toolchain declares it.
#if defined(__has_builtin)
#if __has_builtin(__builtin_amdgcn_global_load_async_to_lds_b128)
#define USE_ASYNC_LDS 1
#endif
#endif

// ---------------------------------------------------------------------------
// K0: zero the accumulator (d_out) -- must happen every call (graph replay).
// ---------------------------------------------------------------------------
__global__ void gcn_zero(float* __restrict__ p, size_t n) {
    size_t i = (size_t)blockIdx.x * blockDim.x + threadIdx.x;
    size_t stride = (size_t)gridDim.x * blockDim.x;
    for (; i < n; i += stride) p[i] = 0.0f;
}

// ---------------------------------------------------------------------------
// K1: m[i,:] = (h[i,:] @ W) * norm[i]   via V_WMMA_F32_16X16X4_F32
// Block = 256 threads = 8 waves; block owns a 16-row tile of h (one
// contiguous 8 KB slab, staged to LDS via async global->LDS when available);
// wave w computes the 16x16 output tile at columns [16w, 16w+16).
// A-frag (16x4 f32): lanes 0-15 -> K=k+0(x),k+1(y); lanes 16-31 -> K=k+2,k+3
// B-frag (4x16 f32): same K split, N = col0 + lane%16
// C/D (16x16 f32, 8 VGPRs): VGPR r: lanes 0-15 M=r, lanes 16-31 M=r+8
// ---------------------------------------------------------------------------
__global__ __launch_bounds__(256) void gcn_gemm_wmma(
    const float* __restrict__ h, const float* __restrict__ norm,
    const float* __restrict__ W, float* __restrict__ m, int N)
{
    const int row0 = blockIdx.x * 16;
    const int wave = threadIdx.x >> 5;   // 0..7 -> column tile
    const int lane = threadIdx.x & 31;
    const int half = lane >> 4;          // 0: K+0/1, 1: K+2/3
    const int l16  = lane & 15;
    const int col0 = wave * 16;
    const int t    = threadIdx.x;

    __shared__ float hs[16 * D];         // 8 KB staged h tile
    __shared__ float ns[16];             // per-row norm for the epilogue

    // Prefetch the W operand (64 KB, L2-resident across the whole grid).
    __builtin_prefetch(&W[(size_t)(t & 127) * D], 0, 1);

    const bool full = (row0 + 16 <= N);  // block-uniform
    if (full) {
        // hs[i] == h[row0*128 + i]: one contiguous 8 KB copy, 32 B/thread.
        const float* g = h + (size_t)row0 * D + t * 8;
        float* l = &hs[t * 8];
#ifdef USE_ASYNC_LDS
        __builtin_amdgcn_global_load_async_to_lds_b128(
            (b128_t*)g, (b128_t*)l, 0, 0);
        __builtin_amdgcn_global_load_async_to_lds_b128(
            (b128_t*)(g + 4), (b128_t*)(l + 4), 0, 0);
#if __has_builtin(__builtin_amdgcn_s_wait_asynccnt)
        __builtin_amdgcn_s_wait_asynccnt(0);
#else
        asm volatile("s_wait_asynccnt 0" ::: "memory");
#endif
#else
        const float4 v0 = ((const float4*)g)[0];
        const float4 v1 = ((const float4*)g)[1];
        ((float4*)l)[0] = v0;
        ((float4*)l)[1] = v1;
#endif
    } else {
        for (int i = t; i < 16 * D; i += 256) {
            int row = row0 + (i >> 7);
            if (row >= N) row = N - 1;
            hs[i] = h[(size_t)row * D + (i & (D - 1))];
        }
    }
    if (t < 16) {
        int row = row0 + t;
        if (row >= N) row = N - 1;
        ns[t] = norm[row];
    }
    __syncthreads();

    v8f acc = {};
    const int ka = half * 2;             // K offset of this half-wave
    #pragma unroll 4
    for (int k = 0; k < D; k += 4) {
        v2f a, b;
        a.x = hs[l16 * D + k + ka + 0];
        a.y = hs[l16 * D + k + ka + 1];
        b.x = W[(size_t)(k + ka + 0) * D + col0 + l16];
        b.y = W[(size_t)(k + ka + 1) * D + col0 + l16];
        // (neg_a, A, neg_b, B, c_mod, C, reuse_a, reuse_b)
        acc = __builtin_amdgcn_wmma_f32_16x16x4_f32(
            false, a, false, b, (short)0, acc, false, false);
    }

    // Epilogue: scale each output row by norm[row]; branch-free fast path.
    float* mt = m + (size_t)row0 * D + (size_t)(half * 8) * D + col0 + l16;
    if (full) {
        #pragma unroll
        for (int r = 0; r < 8; ++r) {
            mt[(size_t)r * D] = acc[r] * ns[r + half * 8];
        }
    } else {
        #pragma unroll
        for (int r = 0; r < 8; ++r) {
            int row = row0 + r + half * 8;
            if (row < N) mt[(size_t)r * D] = acc[r] * ns[r + half * 8];
        }
    }
}

// ---------------------------------------------------------------------------
// K2: scatter-sum. One wave32 per edge: lane reads float4 of m[src] (512 B
// per edge, coalesced; m fits in the 192 MB L2), atomically adds into
// out[dst]. Accumulator (51 MB) also lives in L2 -> L2-resident atomics.
// ---------------------------------------------------------------------------
__global__ __launch_bounds__(256) void gcn_scatter(
    const float* __restrict__ m, const int* __restrict__ src,
    const int* __restrict__ dst, float* __restrict__ acc, int E)
{
    int e = blockIdx.x * 8 + (threadIdx.x >> 5);
    if (e >= E) return;
    const int lane = threadIdx.x & 31;
    const int s = src[e];
    const int d = dst[e];
    const float4 v = ((const float4*)(m + (size_t)s * D))[lane];
    float* o = acc + (size_t)d * D + lane * 4;
    atomicAdd(o + 0, v.x);
    atomicAdd(o + 1, v.y);
    atomicAdd(o + 2, v.z);
    atomicAdd(o + 3, v.w);
}

// ---------------------------------------------------------------------------
// K3: out = LeakyReLU(out * norm + bias, 0.2), in place, float4 per lane.
// ---------------------------------------------------------------------------
__global__ __launch_bounds__(256) void gcn_finalize(
    float* __restrict__ out, const float* __restrict__ norm,
    const float* __restrict__ bias, int N)
{
    int idx = blockIdx.x * blockDim.x + threadIdx.x;  // N*32 threads total
    if (idx >= N * 32) return;
    const int node = idx >> 5;
    const int lane = idx & 31;
    const float nv = norm[node];
    float4* p = (float4*)(out + (size_t)node * D) + lane;
    const float4 b = ((const float4*)bias)[lane];
    float4 v = *p;
    v.x = fmaf(v.x, nv, b.x);
    v.y = fmaf(v.y, nv, b.y);
    v.z = fmaf(v.z, nv, b.z);
    v.w = fmaf(v.w, nv, b.w);
    v.x = v.x > 0.0f ? v.x : 0.2f * v.x;
    v.y = v.y > 0.0f ? v.y : 0.2f * v.y;
    v.z = v.z > 0.0f ? v.z : 0.2f * v.z;
    v.w = v.w > 0.0f ? v.w : 0.2f * v.w;
    *p = v;
}

// ---------------------------------------------------------------------------
// Inputs (setup_inputs order): h[N*128] f32, norm[N] f32, weight[128*128] f32,
// bias[128] f32, src[E] i32, dst[E] i32. Output: [N*128] f32.
// Workspace: m = h@W*norm, N*128*4 = 51.2 MB in d_ws.
// ---------------------------------------------------------------------------
extern "C" void kernel_launch(void* const* d_in, const int* in_sizes, int n_in,
                              void* d_out, int out_size, void* d_ws, size_t ws_size,
                              hipStream_t stream) {
    const float* h    = (const float*)d_in[0];
    const float* norm = (const float*)d_in[1];
    const float* W    = (const float*)d_in[2];
    const float* bias = (const float*)d_in[3];
    const int*   src  = (const int*)d_in[4];
    const int*   dst  = (const int*)d_in[5];
    const int N = in_sizes[1];   // norm has N elements
    const int E = in_sizes[4];

    float* m   = (float*)d_ws;
    float* out = (float*)d_out;

    const size_t total = (size_t)N * D;
    gcn_zero<<<2048, 256, 0, stream>>>(out, total);

    const int row_tiles = (N + 15) / 16;
    gcn_gemm_wmma<<<row_tiles, 256, 0, stream>>>(h, norm, W, m, N);

    gcn_scatter<<<(E + 7) / 8, 256, 0, stream>>>(m, src, dst, out, E);

    gcn_finalize<<<(N * 32 + 255) / 256, 256, 0, stream>>>(out, norm, bias, N);
}